// Seeds_Searcher_17600775979092
// MI455X (gfx1250) — compile-verified
//
#include <hip/hip_runtime.h>
#include <hip/hip_fp16.h>
#include <cstdint>

// Problem geometry (fixed by the reference).
#define BB 128
#define HH 240
#define WW 320
#define HWE (HH * WW)          // 76800 pixels per image
#define MAX_ITERS 100
#define SPLIT 4                // workgroups per batch image
#define TPB 1024               // 32 wave32 waves -> fills one WGP
#define CHUNK (HWE / SPLIT)    // 19200 pixels per workgroup

static __device__ __forceinline__ __half2 u32_as_half2(uint32_t u) {
    union { uint32_t u; __half2 h; } c;
    c.u = u;
    return c.h;
}

// ---------------------------------------------------------------------------
// Zero the output (harness poisons d_out with 0xAA before timing).
// ---------------------------------------------------------------------------
__global__ void zero_out_kernel(float* __restrict__ out, int n) {
    int i = blockIdx.x * blockDim.x + threadIdx.x;
    int stride = gridDim.x * blockDim.x;
    for (; i < n; i += stride) out[i] = 0.0f;
}

// ---------------------------------------------------------------------------
// Pack planar f32 (B,2,H,W) gradient field into interleaved half2 (B,H,W).
// 78.6 MB read + 39.3 MB write: ~5 us at 23.3 TB/s, negligible.
// ---------------------------------------------------------------------------
__global__ void pack_gvf_kernel(const float* __restrict__ Mgvf,
                                uint32_t* __restrict__ packed) {
    int idx = blockIdx.x * blockDim.x + threadIdx.x;
    if (idx >= BB * HWE) return;
    int b = idx / HWE;
    int p = idx - b * HWE;
    const float* base = Mgvf + (size_t)b * 2 * HWE;
    float gh = base[p];
    float gw = base[HWE + p];
    __half2 hv = __floats2half2_rn(gh, gw);
    union { __half2 h; uint32_t u; } c;
    c.h = hv;
    packed[idx] = c.u;
}

// ---------------------------------------------------------------------------
// Main kernel: stage the batch's packed gradient field (300 KB) into LDS via
// CDNA5 async global->LDS copies, then run the per-pixel advection loop with
// all bilerp gathers served from LDS. Scatter with HW f32 global atomics.
// ---------------------------------------------------------------------------
template <bool USE_PACKED>
__global__ __launch_bounds__(TPB, 1)
void seeds_kernel(const float* __restrict__ Mnuc,
                  const float* __restrict__ Mgvf,
                  const uint32_t* __restrict__ packed,
                  float* __restrict__ out) {
    extern __shared__ uint32_t sG[];   // HWE half2 entries = 307200 bytes

    const int b    = blockIdx.x / SPLIT;
    const int part = blockIdx.x % SPLIT;
    const int tid  = threadIdx.x;

    if (USE_PACKED) {
        // Async DMA of the 300 KB packed field into LDS (19200 x 16B chunks).
        // Tracked by ASYNCcnt; generic LDS pointer truncates to the 32-bit
        // LDS byte offset the instruction wants.
        const uint32_t* src = packed + (size_t)b * HWE;
        unsigned lds_base = (unsigned)(uintptr_t)(&sG[0]);
        for (int i = tid; i < HWE / 4; i += TPB) {
            unsigned lds_addr = lds_base + (unsigned)i * 16u;
            uint64_t gaddr = (uint64_t)(uintptr_t)src + (uint64_t)i * 16u;
            asm volatile("global_load_async_to_lds_b128 %0, %1, off"
                         :: "v"(lds_addr), "v"(gaddr)
                         : "memory");
        }
#if __has_builtin(__builtin_amdgcn_s_wait_asynccnt)
        __builtin_amdgcn_s_wait_asynccnt(0);
#else
        asm volatile("s_wait_asynccnt 0" ::: "memory");
#endif
    } else {
        // Fallback staging (ws too small): convert f32 planar -> half2 in LDS.
        const float* gh_p = Mgvf + (size_t)b * 2 * HWE;
        const float* gw_p = gh_p + HWE;
        for (int i = tid; i < HWE; i += TPB) {
            __half2 hv = __floats2half2_rn(gh_p[i], gw_p[i]);
            union { __half2 h; uint32_t u; } c;
            c.h = hv;
            sG[i] = c.u;
        }
    }
    __syncthreads();

    const int p_end = part * CHUNK + CHUNK;
    for (int p = part * CHUNK + tid; p < p_end; p += TPB) {
        const bool active = Mnuc[(size_t)b * HWE + p] > 0.0f;
        float h1 = (float)(p / WW);
        float w1 = (float)(p % WW);
        float h0 = -1.0f, w0 = -1.0f;
        bool flag = false;

        if (active) {
#pragma unroll 1
            for (int it = 0; it < MAX_ITERS; ++it) {
                // Once a particle stops moving or flags OOB its state is
                // frozen forever in the reference, so early-exit is exact.
                bool moving = (fabsf(h1 - h0) > 0.5f) || (fabsf(w1 - w0) > 0.5f);
                if (!moving) break;
                bool oob = (h1 > (float)(HH - 1)) || (h1 < 0.0f) ||
                           (w1 > (float)(WW - 1)) || (w1 < 0.0f);
                if (oob) { flag = true; break; }

                float fh1 = floorf(h1);
                float fw1 = floorf(w1);
                int i0 = min(max((int)fh1, 0), HH - 1);
                int j0 = min(max((int)fw1, 0), WW - 1);
                int i1 = min(i0 + 1, HH - 1);
                int j1 = min(j0 + 1, WW - 1);
                float fh = h1 - fh1;
                float fw = w1 - fw1;

                __half2 c00 = u32_as_half2(sG[i0 * WW + j0]);
                __half2 c01 = u32_as_half2(sG[i0 * WW + j1]);
                __half2 c10 = u32_as_half2(sG[i1 * WW + j0]);
                __half2 c11 = u32_as_half2(sG[i1 * WW + j1]);

                float a00 = (1.0f - fh) * (1.0f - fw);
                float a01 = (1.0f - fh) * fw;
                float a10 = fh * (1.0f - fw);
                float a11 = fh * fw;

                float ghv = __low2float(c00)  * a00 + __low2float(c01)  * a01 +
                            __low2float(c10)  * a10 + __low2float(c11)  * a11;
                float gwv = __high2float(c00) * a00 + __high2float(c01) * a01 +
                            __high2float(c10) * a10 + __high2float(c11) * a11;

                float len = sqrtf(ghv * ghv + gwv * gwv);
                float scale = (len > 1.0f) ? (1.0f / fmaxf(len, 1e-20f)) : 1.0f;
                h0 = h1; w0 = w1;
                h1 += ghv * scale;
                w1 += gwv * scale;
            }
        }

        if (active && !flag) {
            int ih = min(max((int)floorf(h1), 0), HH - 1);
            int iw = min(max((int)floorf(w1), 0), WW - 1);
            float* dst = out + (size_t)b * HWE + ih * WW + iw;
            // Lowers to global_atomic_add_f32 (no return -> STOREcnt).
            __hip_atomic_fetch_add(dst, 1.0f, __ATOMIC_RELAXED,
                                   __HIP_MEMORY_SCOPE_AGENT);
        }
    }
}

// ---------------------------------------------------------------------------
extern "C" void kernel_launch(void* const* d_in, const int* in_sizes, int n_in,
                              void* d_out, int out_size, void* d_ws, size_t ws_size,
                              hipStream_t stream) {
    const float* Mnuc = (const float*)d_in[0];   // (B,1,H,W) f32
    const float* Mgvf = (const float*)d_in[1];   // (B,2,H,W) f32
    float* out = (float*)d_out;                  // (B,H,W) f32

    const int n_out = BB * HWE;
    zero_out_kernel<<<(n_out + TPB - 1) / TPB, TPB, 0, stream>>>(out, n_out);

    const size_t packed_bytes = (size_t)BB * HWE * sizeof(uint32_t);
    const size_t lds_bytes = (size_t)HWE * sizeof(uint32_t);   // 307200 B of 320K WGP LDS

    if (ws_size >= packed_bytes) {
        uint32_t* packed = (uint32_t*)d_ws;
        pack_gvf_kernel<<<(BB * HWE + 255) / 256, 256, 0, stream>>>(Mgvf, packed);
        seeds_kernel<true><<<BB * SPLIT, TPB, lds_bytes, stream>>>(
            Mnuc, Mgvf, packed, out);
    } else {
        seeds_kernel<false><<<BB * SPLIT, TPB, lds_bytes, stream>>>(
            Mnuc, Mgvf, nullptr, out);
    }
}